// ROIPooling_73598559584965
// MI455X (gfx1250) — compile-verified
//
#include <hip/hip_runtime.h>

// ROI max-pooling for MI455X (gfx1250).
// Store-bound: 102.8 MB NT b128 stores (~4.4 us floor @ 23.3 TB/s); the 8 MB
// feature map is L2/WGP$-resident (64 ROIs/image reuse). Bins partition the
// ROI pixels (zero intra-ROI reuse), so direct coalesced loads are optimal —
// no LDS staging. WMMA inapplicable (max over a partition is not bilinear).
//
// Block = (img*roi, ph-row): ROI decode (scalar prologue) amortized over the
// 7 pw-bins of a bin-row -> ~7x fewer dynamic scalar instructions than a
// block-per-bin launch, keeping instruction issue well under the store floor.

typedef float v4f __attribute__((ext_vector_type(4)));

#define PH 7
#define PW 7
#define FH 16
#define FW 16
#define FC 512
#define NIMG 16   // B*N = 4*4
#define NROI 64

__global__ __launch_bounds__(128)
void roi_maxpool_gfx1250(const float* __restrict__ fm,
                         const float* __restrict__ rois,
                         float* __restrict__ out)
{
    const unsigned bid = blockIdx.x;             // 0 .. 7*1024-1
    const unsigned ir  = bid & (NIMG * NROI - 1); // img*NROI + roi, 0..1023
    const int      ph  = (int)(bid >> 10);        // 0..6
    const unsigned img = ir >> 6;                 // /NROI

    // ROI coords: block-uniform -> scalar loads.
    const float r0 = rois[ir * 4 + 0];
    const float r1 = rois[ir * 4 + 1];
    const float r2 = rois[ir * 4 + 2];
    const float r3 = rois[ir * 4 + 3];

    // Faithful to reference: truncate 16*coord toward zero (coords >= 0).
    int h0 = (int)(FH * r0);
    int w0 = (int)(FW * r1);
    int h1 = (int)(FH * r2);
    int w1 = (int)(FW * r3);
    h0 = h0 < 0 ? 0 : h0;
    w0 = w0 < 0 ? 0 : w0;
    h1 = h1 > FH ? FH : h1;
    w1 = w1 > FW ? FW : w1;

    const int hs = (h1 - h0) / PH;   // int truncation of region/7
    const int ws = (w1 - w0) / PW;

    // h-bounds of this bin-row. ph<6 spans [h0+ph*hs, h0+(ph+1)*hs); ph==6
    // extends to region end. hs==0: only the last row is populated.
    int hb0, hb1;
    if (hs > 0) { hb0 = h0 + ph * hs; hb1 = (ph == PH - 1) ? h1 : (hb0 + hs); }
    else        { hb0 = h0;           hb1 = (ph == PH - 1) ? h1 : h0; }

    const unsigned ch = threadIdx.x;  // one float4 of channels per lane (128*4=512)
    const float* base = fm + (size_t)img * (FH * FW * FC) + (size_t)ch * 4;
    float* outb = out + (size_t)ir * (PH * PW * FC)
                      + (size_t)ph * (PW * FC) + (size_t)ch * 4;

    const float ninf = -__builtin_inff();

    for (int pw = 0; pw < PW; ++pw) {
        int wb0, wb1;
        if (ws > 0) { wb0 = w0 + pw * ws; wb1 = (pw == PW - 1) ? w1 : (wb0 + ws); }
        else        { wb0 = w0;           wb1 = (pw == PW - 1) ? w1 : w0; }

        v4f acc = (v4f){ninf, ninf, ninf, ninf};

        for (int h = hb0; h < hb1; ++h) {
            const float* row = base + (size_t)h * (FW * FC);
            // gfx1250 global_prefetch_b8: pull the next pixel row of this bin
            // while the current row is max'd (no LOADcnt cost).
            if (h + 1 < hb1)
                __builtin_prefetch(base + (size_t)(h + 1) * (FW * FC)
                                        + (size_t)wb0 * FC, 0, 3);
            for (int w = wb0; w < wb1; ++w) {
                v4f v = *(const v4f*)(row + (size_t)w * FC);  // coalesced b128
                acc = __builtin_elementwise_max(acc, v);
            }
        }

        // Empty bin (only if hs==0 or ws==0) stores -inf, matching the
        // reference's -inf-initialized scatter-max.
        __builtin_nontemporal_store(acc, (v4f*)(outb + (size_t)pw * FC));
    }
}

extern "C" void kernel_launch(void* const* d_in, const int* in_sizes, int n_in,
                              void* d_out, int out_size, void* d_ws, size_t ws_size,
                              hipStream_t stream) {
    const float* fm   = (const float*)d_in[0];   // (4,4,16,16,512) f32
    const float* rois = (const float*)d_in[1];   // (4,4,64,4) f32
    float* out = (float*)d_out;                  // (4,4,64,7,7,512) f32

    const int nblocks = NIMG * NROI * PH;        // 7,168 blocks (ir x ph-row)
    roi_maxpool_gfx1250<<<nblocks, 128, 0, stream>>>(fm, rois, out);
}